// MHA_57123065036933
// MI455X (gfx1250) — compile-verified
//
#include <hip/hip_runtime.h>

// MHA fp32 on gfx1250: all GEMMs via V_WMMA_F32_16X16X4_F32 (exact fp32 matrix path),
// flash-attention phase 2 (logits never hit HBM). Wave32 layouts per CDNA5 ISA 7.12.2.
// B operands staged in LDS as k-row-pair interleaved float2 so every B-fragment is a
// single conflict-free ds_load_b64 (no v_mov packing in the WMMA inner loop).

#define B_SZ   4
#define T_SEQ  2048
#define NH     8
#define HD     64
#define DM     512          // NH*HD
#define NOUT   1536         // 3*DM
#define PSTRIDE 160         // pair-row stride (floats): +160 == +32 mod 64 banks

typedef float v2f __attribute__((ext_vector_type(2)));
typedef float v8f __attribute__((ext_vector_type(8)));

static __device__ __forceinline__ v8f wmma_f32(v2f a, v2f b, v8f c) {
  // D(16x16,f32) += A(16x4,f32) x B(4x16,f32)
  return __builtin_amdgcn_wmma_f32_16x16x4_f32(
      /*neg_a=*/false, a, /*neg_b=*/false, b,
      /*c_mod=*/(short)0, c, /*reuse_a=*/false, /*reuse_b=*/false);
}

// Interleave two row-float4s into pair-layout and store as two float4s.
static __device__ __forceinline__ void store_pair4(float* dst, float4 a0, float4 a1) {
  float4 s0; s0.x = a0.x; s0.y = a1.x; s0.z = a0.y; s0.w = a1.y;
  float4 s1; s1.x = a0.z; s1.y = a1.z; s1.z = a0.w; s1.w = a1.w;
  *(float4*)(dst)     = s0;
  *(float4*)(dst + 4) = s1;
}

// ---------------------------------------------------------------------------
// Phase 1: qkv = x @ W + b, scattered into Q/K/V [B*H, T, 64] workspace.
// Column e of W maps to (h = e/192, dim = (e%192)/3, which = e%3)  — the
// reference reshapes with the size-3 axis innermost.
// Block: 256 thr (8 waves), tile 128(M) x 64(N), K staged 32 at a time in LDS.
// ---------------------------------------------------------------------------
__global__ __launch_bounds__(256) void qkv_proj_kernel(
    const float* __restrict__ X, const float* __restrict__ W,
    const float* __restrict__ bias,
    float* __restrict__ Qw, float* __restrict__ Kw, float* __restrict__ Vw)
{
  __shared__ float Xs[128 * 36];        // pad 32->36: A-frag reads conflict-free
  __shared__ float Wp[16 * PSTRIDE];    // k-pair interleaved W tile (32x64)

  const int tid  = threadIdx.x;
  const int wave = tid >> 5;
  const int lane = tid & 31;
  const int half = lane >> 4;
  const int l16  = lane & 15;

  const int n0 = blockIdx.x * 64;
  const int m0 = blockIdx.y * 128;

  const v8f vzero = {0.f,0.f,0.f,0.f,0.f,0.f,0.f,0.f};
  v8f acc[4];
  #pragma unroll
  for (int j = 0; j < 4; ++j) acc[j] = vzero;

  const int lp = tid >> 4;              // 0..15: pair-row for loader
  const int lg = tid & 15;              // 0..15: 4-col group for loader

  for (int kk = 0; kk < DM; kk += 32) {
    #pragma unroll
    for (int i = 0; i < 4; ++i) {       // X tile: 128 x 32 = 1024 float4 / 4
      const int idx = tid + i * 256;
      const int row = idx >> 3;
      const int c4  = idx & 7;
      *(float4*)&Xs[row * 36 + c4 * 4] =
          *(const float4*)(X + (size_t)(m0 + row) * DM + kk + c4 * 4);
    }
    {                                   // W tile 32x64 -> pair-interleaved
      const float4 a0 = *(const float4*)(W + (size_t)(kk + 2*lp)     * NOUT + n0 + lg*4);
      const float4 a1 = *(const float4*)(W + (size_t)(kk + 2*lp + 1) * NOUT + n0 + lg*4);
      store_pair4(&Wp[lp * PSTRIDE + lg * 8], a0, a1);
    }
    __syncthreads();

    const float* xrow = &Xs[(wave * 16 + l16) * 36 + half * 2];
    #pragma unroll
    for (int c = 0; c < 8; ++c) {       // 8 K-chunks of 4
      const v2f a = *(const v2f*)(xrow + c * 4);
      const int p = c * 2 + half;       // pair row = (c*4 + 2*half)/2
      #pragma unroll
      for (int j = 0; j < 4; ++j) {
        const v2f b = *(const v2f*)&Wp[p * PSTRIDE + (j * 16 + l16) * 2];
        acc[j] = wmma_f32(a, b, acc[j]);
      }
    }
    __syncthreads();
  }

  // Epilogue: bias + scatter into Q/K/V [(b*8+h)*T + t]*64 + dim
  const int rbase = m0 + wave * 16 + half * 8;
  #pragma unroll
  for (int j = 0; j < 4; ++j) {
    const int e     = n0 + j * 16 + l16;
    const int h     = e / 192;
    const int rem   = e - h * 192;
    const int dim   = rem / 3;
    const int which = rem - dim * 3;
    float* base = (which == 0) ? Qw : ((which == 1) ? Kw : Vw);
    const float bv = bias[e];
    #pragma unroll
    for (int r = 0; r < 8; ++r) {
      const int mrow = rbase + r;                 // global row in [0, B*T)
      const int b_   = mrow >> 11;                // / T_SEQ
      const int t_   = mrow & (T_SEQ - 1);
      base[((size_t)(b_ * NH + h) * T_SEQ + t_) * HD + dim] = acc[j][r] + bv;
    }
  }
}

// ---------------------------------------------------------------------------
// Phase 2: flash attention. Block = 128 q-rows of one (b,h); 8 waves, each
// wave owns 16 q-rows. Loop over 32-key tiles staged in LDS. Online softmax
// with running (max, sum) per row; mask applied as +(m-1)*10000 per reference.
// ---------------------------------------------------------------------------
__global__ __launch_bounds__(256) void attn_kernel(
    const float* __restrict__ Qw, const float* __restrict__ Kw,
    const float* __restrict__ Vw, const float* __restrict__ mask,
    float* __restrict__ out)
{
  __shared__ float Ks[32 * 68];         // pad: B-frag reads conflict-free
  __shared__ float Vp[16 * PSTRIDE];    // key-pair interleaved V tile (32x64)
  __shared__ float Ps[8][16 * 36];      // per-wave P transpose staging

  const int tid  = threadIdx.x;
  const int wave = tid >> 5;
  const int lane = tid & 31;
  const int half = lane >> 4;
  const int l16  = lane & 15;

  const int q0 = blockIdx.x * 128;
  const int bh = blockIdx.y;
  const int b  = bh >> 3;
  const int h  = bh & 7;

  // Resident Q fragments: A-layout, lane = q-row (M), vgpr pair = K split.
  v2f qf[16];
  const float* qbase =
      Qw + ((size_t)bh * T_SEQ + q0 + wave * 16 + l16) * HD + half * 2;
  #pragma unroll
  for (int c = 0; c < 16; ++c) qf[c] = *(const v2f*)(qbase + c * 4);

  const v8f vzero = {0.f,0.f,0.f,0.f,0.f,0.f,0.f,0.f};
  v8f o[4];
  #pragma unroll
  for (int j = 0; j < 4; ++j) o[j] = vzero;
  float mrow[8], lrow[8];
  #pragma unroll
  for (int r = 0; r < 8; ++r) { mrow[r] = -1e30f; lrow[r] = 0.0f; }

  const float* mbase = mask + (size_t)b * T_SEQ * T_SEQ;
  const int qlb = q0 + wave * 16 + half * 8;      // q-row of vgpr index r is qlb+r

  const int lp = tid >> 4;              // loader pair-row / row helpers
  const int lg = tid & 15;

  for (int kt = 0; kt < T_SEQ; kt += 32) {
    #pragma unroll
    for (int i = 0; i < 2; ++i) {       // K tile: 32 x 64, row-major + pad
      const int idx = tid + i * 256;
      const int row = idx >> 4;
      const int c4  = idx & 15;
      *(float4*)&Ks[row * 68 + c4 * 4] =
          *(const float4*)(Kw + ((size_t)bh * T_SEQ + kt + row) * HD + c4 * 4);
    }
    {                                   // V tile 32x64 -> key-pair interleaved
      const size_t r0 = ((size_t)bh * T_SEQ + kt + 2*lp) * HD + lg * 4;
      const float4 a0 = *(const float4*)(Vw + r0);
      const float4 a1 = *(const float4*)(Vw + r0 + HD);
      store_pair4(&Vp[lp * PSTRIDE + lg * 8], a0, a1);
    }
    __syncthreads();

    // S = Q (16x64) x K^T (64x32)  -> two 16x16 C-frags
    v8f s[2]; s[0] = vzero; s[1] = vzero;
    #pragma unroll
    for (int c = 0; c < 16; ++c) {
      const int kb = c * 4 + half * 2;
      #pragma unroll
      for (int t2 = 0; t2 < 2; ++t2) {
        const int key = t2 * 16 + l16;
        const v2f bf = *(const v2f*)(&Ks[key * 68 + kb]);
        s[t2] = wmma_f32(qf[c], bf, s[t2]);
      }
    }

    // Scale + mask + online softmax; stage P (A-layout transpose) in LDS.
    float alpha[8];
    float* Pw = &Ps[wave][0];
    #pragma unroll
    for (int r = 0; r < 8; ++r) {
      const int qrow = qlb + r;
      const float* mp = mbase + (size_t)qrow * T_SEQ + kt;
      float s0 = s[0][r] * 0.125f + (mp[l16]      - 1.0f) * 10000.0f;
      float s1 = s[1][r] * 0.125f + (mp[16 + l16] - 1.0f) * 10000.0f;
      float tmax = fmaxf(s0, s1);
      #pragma unroll
      for (int off = 8; off >= 1; off >>= 1)      // reduce within 16-lane group
        tmax = fmaxf(tmax, __shfl_xor(tmax, off, 32));
      const float mnew = fmaxf(mrow[r], tmax);
      const float al   = __expf(mrow[r] - mnew);
      const float p0   = __expf(s0 - mnew);
      const float p1   = __expf(s1 - mnew);
      float psum = p0 + p1;
      #pragma unroll
      for (int off = 8; off >= 1; off >>= 1)
        psum += __shfl_xor(psum, off, 32);
      lrow[r]  = lrow[r] * al + psum;
      mrow[r]  = mnew;
      alpha[r] = al;
      Pw[(r + 8 * half) * 36 + l16]      = p0;
      Pw[(r + 8 * half) * 36 + 16 + l16] = p1;
    }

    #pragma unroll
    for (int j = 0; j < 4; ++j)
      #pragma unroll
      for (int r = 0; r < 8; ++r)
        o[j][r] *= alpha[r];

    // Same-wave DS RAW: DS ops are in-order per wave; fence codegen + DScnt.
    asm volatile("s_wait_dscnt 0" ::: "memory");

    // O += P (16x32) x V (32x64)
    #pragma unroll
    for (int c = 0; c < 8; ++c) {
      const int kb = c * 4 + half * 2;
      const v2f af = *(const v2f*)(&Pw[l16 * 36 + kb]);
      const int p  = c * 2 + half;
      #pragma unroll
      for (int j = 0; j < 4; ++j) {
        const v2f bf = *(const v2f*)&Vp[p * PSTRIDE + (j * 16 + l16) * 2];
        o[j] = wmma_f32(af, bf, o[j]);
      }
    }
    __syncthreads();   // before next K/V tile overwrite
  }

  // Epilogue: divide by running sum, write out[b, t, h*64 + dim]
  #pragma unroll
  for (int r = 0; r < 8; ++r) mrow[r] = 1.0f / lrow[r];
  #pragma unroll
  for (int j = 0; j < 4; ++j)
    #pragma unroll
    for (int r = 0; r < 8; ++r) {
      const int qrow = qlb + r;
      out[((size_t)(b * T_SEQ + qrow)) * DM + h * HD + j * 16 + l16] =
          o[j][r] * mrow[r];
    }
}

// ---------------------------------------------------------------------------
extern "C" void kernel_launch(void* const* d_in, const int* in_sizes, int n_in,
                              void* d_out, int out_size, void* d_ws, size_t ws_size,
                              hipStream_t stream) {
  (void)in_sizes; (void)n_in; (void)out_size; (void)ws_size;
  const float* x    = (const float*)d_in[0];
  const float* mask = (const float*)d_in[1];
  const float* W    = (const float*)d_in[2];
  const float* bias = (const float*)d_in[3];
  float* out = (float*)d_out;

  const size_t per = (size_t)B_SZ * NH * T_SEQ * HD;   // 4,194,304 floats
  float* Qw = (float*)d_ws;
  float* Kw = Qw + per;
  float* Vw = Kw + per;

  dim3 g1(NOUT / 64, (B_SZ * T_SEQ) / 128);            // (24, 64)
  qkv_proj_kernel<<<g1, dim3(256), 0, stream>>>(x, W, bias, Qw, Kw, Vw);

  dim3 g2(T_SEQ / 128, B_SZ * NH);                     // (16, 32)
  attn_kernel<<<g2, dim3(256), 0, stream>>>(Qw, Kw, Vw, mask, out);
}